// SlotDeltaPredictor_86887188398362
// MI455X (gfx1250) — compile-verified
//
#include <hip/hip_runtime.h>

// SlotDeltaPredictor for MI455X (gfx1250, wave32).
// Persistent kernel: 64 blocks x 256 threads (8 waves); each wave owns one
// batch's 126-step recurrent chain; slots live in LDS; all GEMMs use
// V_WMMA_F32_16X16X4_F32 (full fp32 precision; the workload is latency-bound
// on the recurrence, not tensor-rate-bound). Small weights are staged in LDS
// (shared by the 8 waves); dec_w (256KB) streams from L2. Anti-hoist is done
// with opaque *integer offsets* (not pointers) so LDS accesses keep
// addrspace(3) provenance and lower to DS ops, and dec_w loads stay
// global_load (no FLAT/DScnt coupling).

#define STEPS   126
#define TFRAMES 128
#define BPB     8      // batches per block (= waves per block)

typedef float v2f __attribute__((ext_vector_type(2)));
typedef float v8f __attribute__((ext_vector_type(8)));

// ---- LDS layout (floats) -------------------------------------------------
#define W_CONV 0        // 1024
#define W_TS   1024     // 1024
#define W_Q    2048     // 1024
#define W_K    3072     // 1024
#define W_V    4096     // 1024
#define W_GI   5120     // 3072
#define W_GH   8192     // 3072
#define B_CONV 11264    // 32
#define B_TS   11296    // 32
#define B_GI   11328    // 96
#define B_GH   11424    // 96
#define B_DEC  11520    // 512
#define DW_OFF 12032    // 512 (deconv weights)
#define DEC_OFF 12544   // 8*512 decoder output
#define BATCH0 16640    // per-batch regions
#define PB     2944     // per-batch floats
//   per-batch offsets:
#define XO 0            // 512 : conv-input X (holds curr pixels)
#define FO 512          // 512 : feat / slot-feat (dead after k,v)
#define QO 512          // 128 : q   (overlaps dead feat)
#define AO 640          // 64  : attn
#define UO 704          // 128 : updates
#define KO 1024         // 512
#define VO 1536         // 512
#define SO 2048         // 128 : slots (flat = decoder A row)
#define GIO 2176        // 384
#define GHO 2560        // 384  -> 2944
// total = 16640 + 8*2944 = 40192 floats = 160768 bytes

__device__ __forceinline__ v8f wmma4(v2f a, v2f b, v8f c) {
  return __builtin_amdgcn_wmma_f32_16x16x4_f32(false, a, false, b, (short)0, c,
                                               false, false);
}

// D(16 x NTILES*16) = A(MROWS x K, zero-padded to M=16) * B + bias, *oscale, act.
//  BMODE 0: B[kk][col] = W[col*wstride + kk]  (row-major [N][K]; 8B frag load)
//  BMODE 1: B[kk][col] = W[kk*wstride + col]  (row-major [K][N])
template <int MROWS, int NTILES, int KSTEPS, int ACT, int BMODE>
__device__ __forceinline__ void mm_tile(const float* A, int astride,
                                        const float* W, int wstride,
                                        const float* bias,
                                        float* O, int ostride, int orows,
                                        float oscale, int lane) {
  const int m16 = lane & 15;
  const int kb = (lane >> 4) << 1;
  const int rowbase = (lane >> 4) << 3;

  v8f acc[NTILES];
#pragma unroll
  for (int nt = 0; nt < NTILES; ++nt) {
    const float bv = bias ? bias[nt * 16 + m16] : 0.0f;
#pragma unroll
    for (int r = 0; r < 8; ++r) acc[nt][r] = bv;
  }

#pragma unroll
  for (int ks = 0; ks < KSTEPS; ++ks) {
    const int kk = ks * 4 + kb;
    v2f a;
    if (MROWS >= 16) {
      a = *(const v2f*)(A + m16 * astride + kk);
    } else {
      if (m16 < MROWS) a = *(const v2f*)(A + m16 * astride + kk);
      else { a.x = 0.0f; a.y = 0.0f; }
    }
#pragma unroll
    for (int nt = 0; nt < NTILES; ++nt) {
      v2f b;
      if (BMODE == 0) {
        b = *(const v2f*)(W + (nt * 16 + m16) * wstride + kk);
      } else {
        b.x = W[kk * wstride + nt * 16 + m16];
        b.y = W[(kk + 1) * wstride + nt * 16 + m16];
      }
      acc[nt] = wmma4(a, b, acc[nt]);
    }
  }

#pragma unroll
  for (int nt = 0; nt < NTILES; ++nt) {
#pragma unroll
    for (int r = 0; r < 8; ++r) {
      const int row = rowbase + r;
      if (row < orows) {
        float x = acc[nt][r] * oscale;
        if (ACT == 1)  // exact GELU
          x = 0.5f * x * (1.0f + erff(x * 0.70710678118654752f));
        O[row * ostride + nt * 16 + m16] = x;
      }
    }
  }
}

extern "C" __global__ void __launch_bounds__(256)
slotdelta_kernel(const float* __restrict__ frames,
                 const float* __restrict__ slot_mu,
                 const float* __restrict__ conv_w,
                 const float* __restrict__ conv_b,
                 const float* __restrict__ to_slot_w,
                 const float* __restrict__ to_slot_b,
                 const float* __restrict__ q_w,
                 const float* __restrict__ k_w,
                 const float* __restrict__ v_w,
                 const float* __restrict__ gru_wih,
                 const float* __restrict__ gru_whh,
                 const float* __restrict__ gru_bih,
                 const float* __restrict__ gru_bhh,
                 const float* __restrict__ dec_w,
                 const float* __restrict__ dec_b,
                 const float* __restrict__ deconv_w,
                 const float* __restrict__ deconv_b,
                 float* __restrict__ out) {
  extern __shared__ float lds[];
  const int tid = threadIdx.x;
  const int wave = tid >> 5;
  const int lane = tid & 31;
  const int batch = blockIdx.x * BPB + wave;

  // ---- stage small weights + biases into LDS (shared by all 8 waves)
  for (int i = tid; i < 1024; i += 256) {
    lds[W_CONV + i] = conv_w[i];
    lds[W_TS + i] = to_slot_w[i];
    lds[W_Q + i] = q_w[i];
    lds[W_K + i] = k_w[i];
    lds[W_V + i] = v_w[i];
  }
  for (int i = tid; i < 3072; i += 256) {
    lds[W_GI + i] = gru_wih[i];
    lds[W_GH + i] = gru_whh[i];
  }
  for (int i = tid; i < 512; i += 256) {
    lds[B_DEC + i] = dec_b[i];
    lds[DW_OFF + i] = deconv_w[i];
  }
  if (tid < 32) { lds[B_CONV + tid] = conv_b[tid]; lds[B_TS + tid] = to_slot_b[tid]; }
  if (tid < 96) { lds[B_GI + tid] = gru_bih[tid]; lds[B_GH + tid] = gru_bhh[tid]; }

  {
    float* mybase = lds + BATCH0 + wave * PB;
    for (int i = lane; i < 128; i += 32) mybase[SO + i] = slot_mu[i];
  }
  __syncthreads();

  const float scale = 0.17677669529663687f;  // 1/sqrt(32)
  const float dcb = deconv_b[0];
  const float* fbase = frames + (size_t)batch * TFRAMES * 256;
  float* obase = out + (size_t)batch * STEPS * 256;

#pragma unroll 1
  for (int s = 0; s < STEPS; ++s) {
    // Opaque zero offsets: addresses look loop-variant (no cross-step
    // hoisting / register caching of weight fragments), but pointer
    // provenance is preserved -> LDS stays on the DS path, dec_w stays
    // on the global_load path.
    unsigned lz = 0, dz = 0;
    asm volatile("" : "+v"(lz), "+s"(dz));
    float* L = lds + lz;
    const float* dwp = dec_w + dz;
    float* B = L + BATCH0 + wave * PB;

    // ---- stage curr/prev frame into conv-patch layout X[n=4i+j][c*16+p*4+q]
    const float* curr = fbase + (size_t)(s + 1) * 256;
    const float* prev = fbase + (size_t)s * 256;
#pragma unroll
    for (int e = 0; e < 8; ++e) {
      const int pp = lane * 8 + e;  // contiguous 32B/lane
      const int r = pp >> 4, cc = pp & 15;
      const int n = (r >> 2) * 4 + (cc >> 2);
      const int cpq = (r & 3) * 4 + (cc & 3);
      B[XO + n * 32 + cpq] = curr[pp];
      B[XO + n * 32 + 16 + cpq] = prev[pp];
    }

    // ---- encoder: conv(+GELU), to_slot (in place), k, v
    mm_tile<16, 2, 8, 1, 0>(B + XO, 32, L + W_CONV, 32, L + B_CONV, B + FO, 32, 16, 1.0f, lane);
    mm_tile<16, 2, 8, 0, 0>(B + FO, 32, L + W_TS, 32, L + B_TS, B + FO, 32, 16, 1.0f, lane);
    mm_tile<16, 2, 8, 0, 0>(B + FO, 32, L + W_K, 32, nullptr, B + KO, 32, 16, 1.0f, lane);
    mm_tile<16, 2, 8, 0, 0>(B + FO, 32, L + W_V, 32, nullptr, B + VO, 32, 16, 1.0f, lane);

    // ---- 3 slot-attention iterations (M=4 rows zero-padded to 16)
#pragma unroll 1
    for (int it = 0; it < 3; ++it) {
      unsigned lzi = 0;
      asm volatile("" : "+v"(lzi));  // no hoisting across the 3 iterations
      float* Li = lds + lzi;
      float* Bi = Li + BATCH0 + wave * PB;

      mm_tile<4, 2, 8, 0, 0>(Bi + SO, 32, Li + W_Q, 32, nullptr, Bi + QO, 32, 4, 1.0f, lane);
      // attn = q @ k^T * scale : B-operand = kbuf as row-major [N=16][K=32]
      mm_tile<4, 1, 8, 0, 0>(Bi + QO, 32, Bi + KO, 32, nullptr, Bi + AO, 16, 4, scale, lane);
      // softmax over the 4 slots per position (16 columns -> lanes 0..15)
      if (lane < 16) {
        float a0 = Bi[AO + lane], a1 = Bi[AO + 16 + lane];
        float a2 = Bi[AO + 32 + lane], a3 = Bi[AO + 48 + lane];
        float mx = fmaxf(fmaxf(a0, a1), fmaxf(a2, a3));
        float e0 = __expf(a0 - mx), e1 = __expf(a1 - mx);
        float e2 = __expf(a2 - mx), e3 = __expf(a3 - mx);
        float inv = 1.0f / (e0 + e1 + e2 + e3);
        Bi[AO + lane] = e0 * inv; Bi[AO + 16 + lane] = e1 * inv;
        Bi[AO + 32 + lane] = e2 * inv; Bi[AO + 48 + lane] = e3 * inv;
      }
      // updates = attn @ v : B-operand = vbuf row-major [K=16][N=32]
      mm_tile<4, 2, 4, 0, 1>(Bi + AO, 16, Bi + VO, 32, nullptr, Bi + UO, 32, 4, 1.0f, lane);
      // GRU gates
      mm_tile<4, 6, 8, 0, 0>(Bi + UO, 32, Li + W_GI, 32, Li + B_GI, Bi + GIO, 96, 4, 1.0f, lane);
      mm_tile<4, 6, 8, 0, 0>(Bi + SO, 32, Li + W_GH, 32, Li + B_GH, Bi + GHO, 96, 4, 1.0f, lane);
#pragma unroll
      for (int j = 0; j < 4; ++j) {  // lane = feature d, j = slot
        float ir = Bi[GIO + j * 96 + lane], iz = Bi[GIO + j * 96 + 32 + lane];
        float in_ = Bi[GIO + j * 96 + 64 + lane];
        float hr = Bi[GHO + j * 96 + lane], hz = Bi[GHO + j * 96 + 32 + lane];
        float hn = Bi[GHO + j * 96 + 64 + lane];
        float rg = 1.0f / (1.0f + __expf(-(ir + hr)));
        float zg = 1.0f / (1.0f + __expf(-(iz + hz)));
        float ng = tanhf(in_ + rg * hn);
        float h = Bi[SO + j * 32 + lane];
        Bi[SO + j * 32 + lane] = (1.0f - zg) * ng + zg * h;
      }
    }

    // ---- decoder GEMM, cooperative: A = 8 batches' flat slots (8x128 pad 16),
    //      B = dec_w^T (128x512); each wave computes 4 of 32 N-tiles.
    __syncthreads();
#pragma unroll 1
    for (int i = 0; i < 4; ++i) {
      const int nb = (wave * 4 + i) * 16;
      mm_tile<8, 1, 32, 0, 0>(L + BATCH0 + SO, PB, dwp + nb * 128, 128,
                              L + B_DEC + nb, L + DEC_OFF + nb, 512, 8, 1.0f, lane);
    }
    __syncthreads();

    // ---- deconv (4x4 upsample) + tanh + residual clip + store
    const float* myDec = L + DEC_OFF + wave * 512;
    const float* dwl = L + DW_OFF;
#pragma unroll
    for (int e = 0; e < 8; ++e) {
      const int pp = lane * 8 + e;
      const int r = pp >> 4, cc = pp & 15;
      const int i4 = r >> 2, k4 = r & 3, j4 = cc >> 2, l4 = cc & 3;
      float acc = dcb;
#pragma unroll
      for (int c = 0; c < 32; ++c)
        acc += myDec[c * 16 + i4 * 4 + j4] * dwl[c * 16 + k4 * 4 + l4];
      const float delta = tanhf(acc);
      const float cur = B[XO + (i4 * 4 + j4) * 32 + k4 * 4 + l4];
      obase[(size_t)s * 256 + pp] = fminf(fmaxf(cur + delta, 0.0f), 1.0f);
    }
  }
}

extern "C" void kernel_launch(void* const* d_in, const int* in_sizes, int n_in,
                              void* d_out, int out_size, void* d_ws, size_t ws_size,
                              hipStream_t stream) {
  (void)in_sizes; (void)n_in; (void)out_size; (void)d_ws; (void)ws_size;
  const float* frames    = (const float*)d_in[0];
  const float* slot_mu   = (const float*)d_in[1];
  const float* conv_w    = (const float*)d_in[2];
  const float* conv_b    = (const float*)d_in[3];
  const float* to_slot_w = (const float*)d_in[4];
  const float* to_slot_b = (const float*)d_in[5];
  const float* q_w       = (const float*)d_in[6];
  const float* k_w       = (const float*)d_in[7];
  const float* v_w       = (const float*)d_in[8];
  const float* gru_wih   = (const float*)d_in[9];
  const float* gru_whh   = (const float*)d_in[10];
  const float* gru_bih   = (const float*)d_in[11];
  const float* gru_bhh   = (const float*)d_in[12];
  const float* dec_w     = (const float*)d_in[13];
  const float* dec_b     = (const float*)d_in[14];
  const float* deconv_w  = (const float*)d_in[15];
  const float* deconv_b  = (const float*)d_in[16];
  float* out = (float*)d_out;

  const size_t shmem = (size_t)(BATCH0 + BPB * PB) * sizeof(float);  // 160768 B
  slotdelta_kernel<<<dim3(512 / BPB), dim3(256), shmem, stream>>>(
      frames, slot_mu, conv_w, conv_b, to_slot_w, to_slot_b, q_w, k_w, v_w,
      gru_wih, gru_whh, gru_bih, gru_bhh, dec_w, dec_b, deconv_w, deconv_b, out);
}